// DRL4TSP_32538672235068
// MI455X (gfx1250) — compile-verified
//
#include <hip/hip_runtime.h>
#include <hip/hip_bf16.h>
#include <math.h>

#define BATCH 512
#define HDIM  128
#define SEQ   128
#define FIN   2
#define BT    16
#define NBLK  (BATCH / BT)

typedef __attribute__((ext_vector_type(2))) float v2f;
typedef __attribute__((ext_vector_type(8))) float v8f;

__device__ __forceinline__ v8f wmma4(v2f a, v2f b, v8f c) {
  // D = A(16x4,f32) * B(4x16,f32) + C(16x16,f32)
  return __builtin_amdgcn_wmma_f32_16x16x4_f32(false, a, false, b, (short)0, c,
                                               false, false);
}
__device__ __forceinline__ float sigmoidf_(float x) {
  return 1.0f / (1.0f + __expf(-x));
}

// ---------------------------------------------------------------------------
// Precompute: static_h [b][s][h], A_pre = W1@sh + W2@dh, P_pre = P1@sh
// One block per (s, 16-batch tile). WMMA f32 16x16x4.
// ---------------------------------------------------------------------------
__global__ void __launch_bounds__(256) drl_pre(
    const float* __restrict__ stat, const float* __restrict__ dyn,
    const float* __restrict__ w_s, const float* __restrict__ b_s,
    const float* __restrict__ w_d, const float* __restrict__ b_d,
    const float* __restrict__ attn_W, const float* __restrict__ ptr_W,
    float* __restrict__ staticH, float* __restrict__ A_pre,
    float* __restrict__ P_pre) {
  __shared__ float shT[HDIM * 17];  // [k=h][b] padded
  __shared__ float dhT[HDIM * 17];
  __shared__ float oA[BT * 132];    // [b][h] padded (store transpose)
  __shared__ float oP[BT * 132];

  const int tid = threadIdx.x;
  const int lane = tid & 31;
  const int w = tid >> 5;        // wave 0..7 -> h-tile
  const int l16 = lane & 15;
  const int lh = lane >> 4;      // half-wave
  const int s = blockIdx.x & (SEQ - 1);
  const int b0 = (blockIdx.x >> 7) * BT;

  for (int idx = tid; idx < BT * HDIM; idx += 256) {
    int h = idx & (HDIM - 1);
    int b = idx >> 7;
    int bg = b0 + b;
    size_t base = (size_t)bg * FIN * SEQ;
    float s0v = stat[base + s];
    float s1v = stat[base + SEQ + s];
    float d0v = dyn[base + s];
    float d1v = dyn[base + SEQ + s];
    float sv = w_s[h * 2] * s0v + w_s[h * 2 + 1] * s1v + b_s[h];
    float dv = w_d[h * 2] * d0v + w_d[h * 2 + 1] * d1v + b_d[h];
    shT[h * 17 + b] = sv;
    dhT[h * 17 + b] = dv;
    staticH[((size_t)bg * SEQ + s) * HDIM + h] = sv;
  }
  __syncthreads();

  v8f accA = {};
  v8f accP = {};
  const int rowW = w * 16 + l16;
  for (int k = 0; k < HDIM; k += 4) {
    int kc = k + lh * 2;
    v2f bs;
    bs.x = shT[kc * 17 + l16];
    bs.y = shT[(kc + 1) * 17 + l16];
    v2f bd2;
    bd2.x = dhT[kc * 17 + l16];
    bd2.y = dhT[(kc + 1) * 17 + l16];
    const float* w1 = attn_W + (size_t)rowW * 384 + kc;  // W1 = attn_W[:,0:H]
    v2f a;
    a.x = w1[0];
    a.y = w1[1];
    accA = wmma4(a, bs, accA);
    a.x = w1[128];                                       // W2 = attn_W[:,H:2H]
    a.y = w1[129];
    accA = wmma4(a, bd2, accA);
    const float* p1 = ptr_W + (size_t)rowW * 256 + kc;   // P1 = ptr_W[:,0:H]
    a.x = p1[0];
    a.y = p1[1];
    accP = wmma4(a, bs, accP);
  }
#pragma unroll
  for (int v = 0; v < 8; ++v) {
    int h = w * 16 + v + lh * 8;
    oA[l16 * 132 + h] = accA[v];
    oP[l16 * 132 + h] = accP[v];
  }
  __syncthreads();
  for (int idx = tid; idx < BT * HDIM; idx += 256) {
    int h = idx & (HDIM - 1);
    int b = idx >> 7;
    size_t g = ((size_t)(b0 + b) * SEQ + s) * HDIM + h;
    A_pre[g] = oA[b * 132 + h];
    P_pre[g] = oP[b * 132 + h];
  }
}

// ---------------------------------------------------------------------------
// Scan: one block (16 waves) owns 16 batch elements for all 128 steps.
// Waves 0-7 ("A"): r/z GEMMs, h state, gates, low-K halves of u/pc.
// Waves 8-15 ("B"): gi_n/gh_n GEMMs, high-K halves of u/pc.
// ---------------------------------------------------------------------------
__global__ void __launch_bounds__(512) drl_scan(
    const float* __restrict__ stat,
    const float* __restrict__ w_dec, const float* __restrict__ b_dec,
    const float* __restrict__ w_ih, const float* __restrict__ w_hh,
    const float* __restrict__ b_ih, const float* __restrict__ b_hh,
    const float* __restrict__ attn_v, const float* __restrict__ attn_W,
    const float* __restrict__ ptr_v, const float* __restrict__ ptr_W,
    const float* __restrict__ staticH, const float* __restrict__ A_pre,
    const float* __restrict__ P_pre, float* __restrict__ out) {
  __shared__ float xh[256 * 17];   // [k][b]: k<128 -> x, k>=128 -> h
  __shared__ float uL[BT * 132];   // [b][h] (also pc partial-lo)
  __shared__ float pcL[BT * 132];  // [b][h] (also u partial-hi)
  __shared__ float ctxT[HDIM * 17];// [k=h][b]; also gh_n staging [h*17+b]
  __shared__ float eL[BT * SEQ];   // energies/attns/logits; also gi_n [h*16+b]
  __shared__ float brz[256];
  __shared__ float binL[128];
  __shared__ float bhnL[128];
  __shared__ float avL[128];
  __shared__ float pvL[128];
  __shared__ float wdL[256];
  __shared__ float bdL[128];
  __shared__ float decL[BT * 2];

  const int tid = threadIdx.x;
  const int lane = tid & 31;
  const int w = tid >> 5;         // 0..15
  const int l16 = lane & 15;
  const int lh = lane >> 4;
  const int b0 = blockIdx.x * BT;
  const bool isA = (w < 8);
  const int wt = isA ? w : (w - 8);      // row-tile index 0..7 for this wave
  const int rowR = wt * 16 + l16;        // weight row within 128-row group

  // one-time block init
  if (tid < 256) {
    brz[tid] = b_ih[tid] + b_hh[tid];
    wdL[tid] = w_dec[tid];
  }
  if (tid < 128) {
    binL[tid] = b_ih[256 + tid];
    bhnL[tid] = b_hh[256 + tid];
    avL[tid] = attn_v[tid];
    pvL[tid] = ptr_v[tid];
    bdL[tid] = b_dec[tid];
  }
  for (int i = tid; i < 128 * 17; i += 512) xh[128 * 17 + i] = 0.0f;  // h0=0
  if (tid < BT * 2) decL[tid] = 0.0f;                                  // dec0
  v8f hreg = {};
  __syncthreads();

  for (int step = 0; step < SEQ; ++step) {
    // x = w_decoder @ dec_in + b_decoder, into xh[k][b] (k<128)
    {
      int k = tid >> 4;
      int b = tid & 15;
      if (tid < HDIM * BT)
        xh[k * 17 + b] = wdL[k * 2] * decL[b * 2] +
                         wdL[k * 2 + 1] * decL[b * 2 + 1] + bdL[k];
    }
    __syncthreads();

    // ---- GRU GEMMs (split across wave groups) ----
    v8f ar = {}, az = {};
    if (isA) {
      for (int k = 0; k < HDIM; k += 4) {
        int kc = k + lh * 2;
        v2f bx;
        bx.x = xh[kc * 17 + l16];
        bx.y = xh[(kc + 1) * 17 + l16];
        const float* wr = w_ih + (size_t)rowR * HDIM + kc;
        v2f a;
        a.x = wr[0];
        a.y = wr[1];
        ar = wmma4(a, bx, ar);
        a.x = wr[128 * HDIM];
        a.y = wr[128 * HDIM + 1];
        az = wmma4(a, bx, az);
      }
      for (int k = 0; k < HDIM; k += 4) {
        int kc = k + lh * 2;
        v2f bh;
        bh.x = xh[(128 + kc) * 17 + l16];
        bh.y = xh[(128 + kc + 1) * 17 + l16];
        const float* wr = w_hh + (size_t)rowR * HDIM + kc;
        v2f a;
        a.x = wr[0];
        a.y = wr[1];
        ar = wmma4(a, bh, ar);
        a.x = wr[128 * HDIM];
        a.y = wr[128 * HDIM + 1];
        az = wmma4(a, bh, az);
      }
    } else {
      v8f agin = {}, aghn = {};
      for (int k = 0; k < HDIM; k += 4) {
        int kc = k + lh * 2;
        v2f bx;
        bx.x = xh[kc * 17 + l16];
        bx.y = xh[(kc + 1) * 17 + l16];
        const float* wn = w_ih + (size_t)(256 + rowR) * HDIM + kc;
        v2f a;
        a.x = wn[0];
        a.y = wn[1];
        agin = wmma4(a, bx, agin);
      }
      for (int k = 0; k < HDIM; k += 4) {
        int kc = k + lh * 2;
        v2f bh;
        bh.x = xh[(128 + kc) * 17 + l16];
        bh.y = xh[(128 + kc + 1) * 17 + l16];
        const float* wn = w_hh + (size_t)(256 + rowR) * HDIM + kc;
        v2f a;
        a.x = wn[0];
        a.y = wn[1];
        aghn = wmma4(a, bh, aghn);
      }
#pragma unroll
      for (int v = 0; v < 8; ++v) {
        int j = wt * 16 + v + lh * 8;
        eL[j * 16 + l16] = agin[v];     // gi_n staging [h][b]
        ctxT[j * 17 + l16] = aghn[v];   // gh_n staging [h][b]
      }
    }
    __syncthreads();  // old h fully read, gi_n/gh_n staged

    // ---- gates (waves A), write h_new to LDS ----
    if (isA) {
#pragma unroll
      for (int v = 0; v < 8; ++v) {
        int j = wt * 16 + v + lh * 8;
        float r = sigmoidf_(ar[v] + brz[j]);
        float z = sigmoidf_(az[v] + brz[128 + j]);
        float n = tanhf(eL[j * 16 + l16] + binL[j] +
                        r * (ctxT[j * 17 + l16] + bhnL[j]));
        hreg[v] = (1.0f - z) * n + z * hreg[v];
        xh[(128 + j) * 17 + l16] = hreg[v];
      }
    }
    __syncthreads();

    // ---- u = W3 @ h_new, K split across wave pairs ----
    {
      v8f au = {};
      const int k0 = isA ? 0 : 64;
      for (int k = 0; k < 64; k += 4) {
        int kc = k0 + k + lh * 2;
        v2f bh;
        bh.x = xh[(128 + kc) * 17 + l16];
        bh.y = xh[(128 + kc + 1) * 17 + l16];
        const float* wv = attn_W + (size_t)rowR * 384 + 256 + kc;
        v2f a;
        a.x = wv[0];
        a.y = wv[1];
        au = wmma4(a, bh, au);
      }
      float* dst = isA ? uL : pcL;
#pragma unroll
      for (int v = 0; v < 8; ++v)
        dst[l16 * 132 + wt * 16 + v + lh * 8] = au[v];
    }
    __syncthreads();

    // ---- e[b,s] = sum_h attn_v[h] * tanh(A_pre[b,s,h] + u[b,h]) ----
#pragma unroll
    for (int p = 0; p < 4; ++p) {
      int idx = p * 512 + tid;
      int b = idx >> 7;
      int s = idx & 127;
      const float* Ap = A_pre + ((size_t)(b0 + b) * SEQ + s) * HDIM;
      const float* u1 = uL + b * 132;
      const float* u2 = pcL + b * 132;
      if (p < 3) __builtin_prefetch(Ap + 4 * SEQ * HDIM, 0, 1);
      float acc = 0.0f;
      for (int h = 0; h < HDIM; h += 4) {
        float4 a4 = *(const float4*)(Ap + h);
        float4 x4 = *(const float4*)(u1 + h);
        float4 y4 = *(const float4*)(u2 + h);
        acc += avL[h] * tanhf(a4.x + x4.x + y4.x);
        acc += avL[h + 1] * tanhf(a4.y + x4.y + y4.y);
        acc += avL[h + 2] * tanhf(a4.z + x4.z + y4.z);
        acc += avL[h + 3] * tanhf(a4.w + x4.w + y4.w);
      }
      eL[b * SEQ + s] = acc;
    }
    __syncthreads();

    // ---- softmax over s (one wave per batch row) ----
    {
      float m = -3.4e38f;
#pragma unroll
      for (int i = 0; i < 4; ++i) m = fmaxf(m, eL[w * SEQ + lane + i * 32]);
      for (int off = 16; off >= 1; off >>= 1)
        m = fmaxf(m, __shfl_xor(m, off, 32));
      float ex[4];
      float ssum = 0.0f;
#pragma unroll
      for (int i = 0; i < 4; ++i) {
        ex[i] = __expf(eL[w * SEQ + lane + i * 32] - m);
        ssum += ex[i];
      }
      for (int off = 16; off >= 1; off >>= 1) ssum += __shfl_xor(ssum, off, 32);
      float inv = 1.0f / ssum;
#pragma unroll
      for (int i = 0; i < 4; ++i) eL[w * SEQ + lane + i * 32] = ex[i] * inv;
    }
    __syncthreads();

    // ---- context[b,h] = sum_s attns[b,s] * static_h[b,s,h] -> ctxT[h][b] --
#pragma unroll
    for (int bb = 0; bb < 4; ++bb) {
      int b = bb * 4 + (tid >> 7);
      int h = tid & 127;
      const float* sh = staticH + (size_t)(b0 + b) * SEQ * HDIM + h;
      float acc = 0.0f;
      for (int s = 0; s < SEQ; ++s)
        acc += eL[b * SEQ + s] * sh[(size_t)s * HDIM];
      ctxT[h * 17 + b] = acc;
    }
    __syncthreads();

    // ---- pc = P2 @ context, K split across wave pairs ----
    {
      v8f apc = {};
      const int k0 = isA ? 0 : 64;
      for (int k = 0; k < 64; k += 4) {
        int kc = k0 + k + lh * 2;
        v2f bc;
        bc.x = ctxT[kc * 17 + l16];
        bc.y = ctxT[(kc + 1) * 17 + l16];
        const float* pw = ptr_W + (size_t)rowR * 256 + 128 + kc;
        v2f a;
        a.x = pw[0];
        a.y = pw[1];
        apc = wmma4(a, bc, apc);
      }
      float* dst = isA ? uL : pcL;
#pragma unroll
      for (int v = 0; v < 8; ++v)
        dst[l16 * 132 + wt * 16 + v + lh * 8] = apc[v];
    }
    __syncthreads();

    // ---- logits[b,s] = sum_h ptr_v[h] * tanh(P_pre[b,s,h] + pc[b,h]) ----
#pragma unroll
    for (int p = 0; p < 4; ++p) {
      int idx = p * 512 + tid;
      int b = idx >> 7;
      int s = idx & 127;
      const float* Pp = P_pre + ((size_t)(b0 + b) * SEQ + s) * HDIM;
      const float* u1 = uL + b * 132;
      const float* u2 = pcL + b * 132;
      float acc = 0.0f;
      for (int h = 0; h < HDIM; h += 4) {
        float4 a4 = *(const float4*)(Pp + h);
        float4 x4 = *(const float4*)(u1 + h);
        float4 y4 = *(const float4*)(u2 + h);
        acc += pvL[h] * tanhf(a4.x + x4.x + y4.x);
        acc += pvL[h + 1] * tanhf(a4.y + x4.y + y4.y);
        acc += pvL[h + 2] * tanhf(a4.z + x4.z + y4.z);
        acc += pvL[h + 3] * tanhf(a4.w + x4.w + y4.w);
      }
      eL[b * SEQ + s] = acc;
    }
    __syncthreads();

    // ---- argmax + logp + output + next dec_in (one wave per row) ----
    {
      float m = -3.4e38f;
      int am = 0;
#pragma unroll
      for (int i = 0; i < 4; ++i) {
        int s = lane + i * 32;
        float vle = eL[w * SEQ + s];
        if (vle > m || (vle == m && s < am)) {
          m = vle;
          am = s;
        }
      }
      for (int off = 16; off >= 1; off >>= 1) {
        float om = __shfl_xor(m, off, 32);
        int oam = __shfl_xor(am, off, 32);
        if (om > m || (om == m && oam < am)) {
          m = om;
          am = oam;
        }
      }
      float ssum = 0.0f;
#pragma unroll
      for (int i = 0; i < 4; ++i)
        ssum += __expf(eL[w * SEQ + lane + i * 32] - m);
      for (int off = 16; off >= 1; off >>= 1) ssum += __shfl_xor(ssum, off, 32);
      if (lane == 0) {
        int bg = b0 + w;
        out[(size_t)bg * SEQ + step] = (float)am;  // tour_idx
        out[(size_t)BATCH * SEQ + (size_t)bg * SEQ + step] =
            -__logf(ssum);                         // logp (logit[am]==max)
        size_t base = (size_t)bg * FIN * SEQ;
        decL[w * 2] = stat[base + am];
        decL[w * 2 + 1] = stat[base + SEQ + am];
      }
    }
    __syncthreads();
  }
}

extern "C" void kernel_launch(void* const* d_in, const int* in_sizes, int n_in,
                              void* d_out, int out_size, void* d_ws,
                              size_t ws_size, hipStream_t stream) {
  const float* stat = (const float*)d_in[0];
  const float* dyn = (const float*)d_in[1];
  const float* w_s = (const float*)d_in[2];
  const float* b_s = (const float*)d_in[3];
  const float* w_d = (const float*)d_in[4];
  const float* b_d = (const float*)d_in[5];
  const float* w_dec = (const float*)d_in[6];
  const float* b_dec = (const float*)d_in[7];
  const float* w_ih = (const float*)d_in[8];
  const float* w_hh = (const float*)d_in[9];
  const float* b_ih = (const float*)d_in[10];
  const float* b_hh = (const float*)d_in[11];
  const float* attn_v = (const float*)d_in[12];
  const float* attn_W = (const float*)d_in[13];
  const float* ptr_v = (const float*)d_in[14];
  const float* ptr_W = (const float*)d_in[15];
  float* out = (float*)d_out;

  float* ws = (float*)d_ws;
  const size_t T = (size_t)BATCH * SEQ * HDIM;  // 8,388,608 floats each
  float* staticH = ws;
  float* A_pre = ws + T;
  float* P_pre = ws + 2 * T;

  drl_pre<<<SEQ * (BATCH / BT), 256, 0, stream>>>(
      stat, dyn, w_s, b_s, w_d, b_d, attn_W, ptr_W, staticH, A_pre, P_pre);
  drl_scan<<<NBLK, 512, 0, stream>>>(stat, w_dec, b_dec, w_ih, w_hh, b_ih,
                                     b_hh, attn_v, attn_W, ptr_v, ptr_W,
                                     staticH, A_pre, P_pre, out);
}